// CopyDecoder_56229711839554
// MI455X (gfx1250) — compile-verified
//
#include <hip/hip_runtime.h>

#define V 50000
#define E 256
#define H 512
#define BS 64
#define SEQ 200
#define XW (E + 2*H)   // 1280
#define H2 (2*H)       // 1024
#define H3 (3*H)       // 1536
#define SPLIT 8        // waves per row tile in fused score_c GEMM (each takes 32/8=4 col tiles)
#define MT ((BS*SEQ)/16) // 800 row tiles of the fused GEMM

typedef __attribute__((ext_vector_type(2))) float v2f;
typedef __attribute__((ext_vector_type(8))) float v8f;

__device__ __forceinline__ v8f wmma_f32_16x16x4(v2f a, v2f b, v8f c) {
  // 8 args: (neg_a, A, neg_b, B, c_mod, C, reuse_a, reuse_b)
  return __builtin_amdgcn_wmma_f32_16x16x4_f32(false, a, false, b, (short)0, c, false, false);
}

// ---------------------------------------------------------------------------
// Prologue: build x = [embed_W[idx], weighted] and effective prev_state.
// Handles the order==0 branch on-device (scalar fallback; dead for order=1).
// ---------------------------------------------------------------------------
__global__ void prologue_kernel(const int* __restrict__ input_idx,
                                const float* __restrict__ encoded,
                                const float* __restrict__ prev_state,
                                const float* __restrict__ weighted,
                                const int* __restrict__ order_p,
                                const float* __restrict__ embed_W,
                                const float* __restrict__ Ws_W,
                                const float* __restrict__ Ws_b,
                                float* __restrict__ x, float* __restrict__ ps) {
  const int b = blockIdx.x;
  const int tid = threadIdx.x;
  const int order = order_p[0];
  const float* e = embed_W + (size_t)input_idx[b] * E;
  float* xb = x + (size_t)b * XW;
  for (int i = tid; i < E; i += 256) xb[i] = e[i];
  if (order != 0) {
    const float* wb = weighted + (size_t)b * H2;
    for (int i = tid; i < H2; i += 256) xb[E + i] = wb[i];
    for (int i = tid; i < H; i += 256) ps[(size_t)b*H + i] = prev_state[(size_t)b*H + i];
  } else {
    for (int i = tid; i < H2; i += 256) xb[E + i] = 0.0f;
    const float* enc1 = encoded + ((size_t)b * SEQ + 1) * H2;
    for (int hh = tid; hh < H; hh += 256) {
      float acc = Ws_b[hh];
      const float* wr = Ws_W + (size_t)hh * H2;
      for (int k = 0; k < H2; ++k) acc += enc1[k] * wr[k];
      ps[(size_t)b*H + hh] = acc;
    }
  }
}

// ---------------------------------------------------------------------------
// Generic C(M,N) = A(M,K) * W(N,K)^T + bias.  K is compile-time so the NTILE
// B-tiles live at constant byte offsets from ONE base pointer (folds into
// the 24-bit instruction offset; no per-tile 64-bit pointer arithmetic).
// One-step software pipeline: next k-step's fragments are loaded into fresh
// registers before this step's WMMAs are issued. All control flow is
// wave-uniform -> EXEC all-ones at every v_wmma.
// ---------------------------------------------------------------------------
template<int NTILE, int K>
__global__ void gemm_bias_wmma(const float* __restrict__ A, const float* __restrict__ W,
                               const float* __restrict__ bias, float* __restrict__ C,
                               int M, int N) {
  const int wpb = blockDim.x >> 5;
  const int wave = blockIdx.x * wpb + (threadIdx.x >> 5);
  const int lane = threadIdx.x & 31;
  const int ntg = (N >> 4) / NTILE;         // groups of NTILE column tiles
  const int total = (M >> 4) * ntg;
  if (wave >= total) return;
  const int tm = wave / ntg, tg = wave % ntg;
  const int half = lane >> 4, l15 = lane & 15;
  const float* Ap = A + (size_t)(tm*16 + l15) * K + half*2;
  const float* Wp = W + (size_t)(tg*NTILE*16 + l15) * K + half*2;  // tile t at +t*16*K

  v8f acc[NTILE];
#pragma unroll
  for (int t = 0; t < NTILE; ++t) acc[t] = (v8f){0.f,0.f,0.f,0.f,0.f,0.f,0.f,0.f};

  v2f a_cur = *(const v2f*)(Ap);
  v2f b_cur[NTILE];
#pragma unroll
  for (int t = 0; t < NTILE; ++t) b_cur[t] = *(const v2f*)(Wp + t*16*K);

#pragma unroll 4
  for (int k = 0; k < K - 4; k += 4) {
    const v2f a_nxt = *(const v2f*)(Ap + k + 4);
    v2f b_nxt[NTILE];
#pragma unroll
    for (int t = 0; t < NTILE; ++t) b_nxt[t] = *(const v2f*)(Wp + t*16*K + k + 4);
#pragma unroll
    for (int t = 0; t < NTILE; ++t) acc[t] = wmma_f32_16x16x4(a_cur, b_cur[t], acc[t]);
    a_cur = a_nxt;
#pragma unroll
    for (int t = 0; t < NTILE; ++t) b_cur[t] = b_nxt[t];
  }
#pragma unroll
  for (int t = 0; t < NTILE; ++t) acc[t] = wmma_f32_16x16x4(a_cur, b_cur[t], acc[t]);

#pragma unroll
  for (int t = 0; t < NTILE; ++t) {
    const int col = (tg*NTILE + t)*16 + l15;
    const float bv = bias[col];
#pragma unroll
    for (int v = 0; v < 8; ++v)
      C[(size_t)(tm*16 + half*8 + v) * N + col] = acc[t][v] + bv;
  }
}

// ---------------------------------------------------------------------------
// GRU gates: state = (1-z)*n + z*prev
// ---------------------------------------------------------------------------
__global__ void gates_kernel(const float* __restrict__ gi, const float* __restrict__ gh,
                             const float* __restrict__ ps,
                             float* __restrict__ state_ws, float* __restrict__ state_out) {
  const int idx = blockIdx.x * blockDim.x + threadIdx.x;  // BS*H
  const int b = idx / H, h = idx % H;
  const float* gib = gi + (size_t)b * H3;
  const float* ghb = gh + (size_t)b * H3;
  const float i_r = gib[h], i_z = gib[H + h], i_n = gib[2*H + h];
  const float h_r = ghb[h], h_z = ghb[H + h], h_n = ghb[2*H + h];
  const float r = 1.0f / (1.0f + __expf(-(i_r + h_r)));
  const float z = 1.0f / (1.0f + __expf(-(i_z + h_z)));
  const float n = tanhf(i_n + r * h_n);
  const float st = (1.0f - z) * n + z * ps[idx];
  state_ws[idx] = st;
  state_out[idx] = st;
}

// ---------------------------------------------------------------------------
// Fused score_c GEMM: each wave owns one 16-row tile and 4 of the 32 column
// tiles; ONE pass over K (compile-time 1024) with 4 accumulators and
// constant-offset B tiles + one-step pipeline. Epilogue: tanh(acc+b) *
// state[b,col], 16-lane shuffle reduce, deterministic split partials.
// ---------------------------------------------------------------------------
__global__ void score_c_gemm(const float* __restrict__ encoded, const float* __restrict__ WcW,
                             const float* __restrict__ Wcb, const float* __restrict__ state,
                             float* __restrict__ partial) {
  const int wpb = blockDim.x >> 5;
  const int gw = blockIdx.x * wpb + (threadIdx.x >> 5);
  const int lane = threadIdx.x & 31;
  if (gw >= MT * SPLIT) return;
  const int tm = gw / SPLIT;
  const int sp = gw % SPLIT;
  const int half = lane >> 4, l15 = lane & 15;
  const int NTC = (H >> 4) / SPLIT;  // 4 column tiles per wave
  const float* Ap = encoded + (size_t)(tm*16 + l15) * H2 + half*2;
  const float* Wp = WcW + (size_t)((sp*NTC)*16 + l15) * H2 + half*2;  // tile t at +t*16*H2

  v8f acc[4];
#pragma unroll
  for (int t = 0; t < 4; ++t) acc[t] = (v8f){0.f,0.f,0.f,0.f,0.f,0.f,0.f,0.f};

  v2f a_cur = *(const v2f*)(Ap);
  v2f b_cur[4];
#pragma unroll
  for (int t = 0; t < 4; ++t) b_cur[t] = *(const v2f*)(Wp + t*16*H2);

#pragma unroll 4
  for (int k = 0; k < H2 - 4; k += 4) {
    const v2f a_nxt = *(const v2f*)(Ap + k + 4);
    v2f b_nxt[4];
#pragma unroll
    for (int t = 0; t < 4; ++t) b_nxt[t] = *(const v2f*)(Wp + t*16*H2 + k + 4);
#pragma unroll
    for (int t = 0; t < 4; ++t) acc[t] = wmma_f32_16x16x4(a_cur, b_cur[t], acc[t]);
    a_cur = a_nxt;
#pragma unroll
    for (int t = 0; t < 4; ++t) b_cur[t] = b_nxt[t];
  }
#pragma unroll
  for (int t = 0; t < 4; ++t) acc[t] = wmma_f32_16x16x4(a_cur, b_cur[t], acc[t]);

  float rowsum[8] = {0.f,0.f,0.f,0.f,0.f,0.f,0.f,0.f};
#pragma unroll
  for (int t = 0; t < 4; ++t) {
    const int col = (sp*NTC + t)*16 + l15;
    const float bv = Wcb[col];
#pragma unroll
    for (int v = 0; v < 8; ++v) {
      const int row = tm*16 + half*8 + v;
      const int bb = row / SEQ;
      rowsum[v] += tanhf(acc[t][v] + bv) * state[(size_t)bb * H + col];
    }
  }
  // reduce across the 16 lanes of each half (lanes 0-15 / 16-31 independently)
#pragma unroll
  for (int v = 0; v < 8; ++v) {
    float s = rowsum[v];
    s += __shfl_xor(s, 8, 32);
    s += __shfl_xor(s, 4, 32);
    s += __shfl_xor(s, 2, 32);
    s += __shfl_xor(s, 1, 32);
    rowsum[v] = s;
  }
  if (l15 == 0) {
#pragma unroll
    for (int v = 0; v < 8; ++v)
      partial[(size_t)(tm * SPLIT + sp) * 16 + half*8 + v] = rowsum[v];
  }
}

__global__ void score_c_finish(const float* __restrict__ partial, const int* __restrict__ enc_idx,
                               float* __restrict__ score_c) {
  const int row = blockIdx.x * blockDim.x + threadIdx.x;
  if (row >= BS * SEQ) return;
  const int tm = row >> 4, r16 = row & 15;
  float s = 0.f;
#pragma unroll
  for (int sp = 0; sp < SPLIT; ++sp) s += partial[(size_t)(tm * SPLIT + sp) * 16 + r16];
  s += (enc_idx[row] == 0) ? -1000.0f : 0.0f;
  score_c[row] = tanhf(s);
}

// ---------------------------------------------------------------------------
// Softmax over [score_g | score_c] per batch row; in place.
// ---------------------------------------------------------------------------
__global__ void softmax_kernel(float* __restrict__ g, float* __restrict__ c) {
  __shared__ float red[256];
  __shared__ float sval;
  const int b = blockIdx.x, tid = threadIdx.x;
  float* gb = g + (size_t)b * V;
  float* cb = c + (size_t)b * SEQ;
  float mx = -3.0e38f;
  for (int i = tid; i < V; i += 256) mx = fmaxf(mx, gb[i]);
  for (int i = tid; i < SEQ; i += 256) mx = fmaxf(mx, cb[i]);
  red[tid] = mx; __syncthreads();
  for (int off = 128; off > 0; off >>= 1) {
    if (tid < off) red[tid] = fmaxf(red[tid], red[tid + off]);
    __syncthreads();
  }
  if (tid == 0) sval = red[0];
  __syncthreads();
  const float m = sval;
  __syncthreads();
  float s = 0.f;
  for (int i = tid; i < V; i += 256) s += __expf(gb[i] - m);
  for (int i = tid; i < SEQ; i += 256) s += __expf(cb[i] - m);
  red[tid] = s; __syncthreads();
  for (int off = 128; off > 0; off >>= 1) {
    if (tid < off) red[tid] += red[tid + off];
    __syncthreads();
  }
  if (tid == 0) sval = 1.0f / red[0];
  __syncthreads();
  const float inv = sval;
  for (int i = tid; i < V; i += 256) gb[i] = __expf(gb[i] - m) * inv;
  for (int i = tid; i < SEQ; i += 256) cb[i] = __expf(cb[i] - m) * inv;
}

// ---------------------------------------------------------------------------
// Deterministic scatter-add of prob_c into the vocab probs (one thread per b).
// ---------------------------------------------------------------------------
__global__ void scatter_kernel(const float* __restrict__ prob_c, const int* __restrict__ enc_idx,
                               float* __restrict__ out) {
  const int b = threadIdx.x;
  if (b >= BS) return;
  float* ob = out + (size_t)b * V;
  const float* pc = prob_c + (size_t)b * SEQ;
  const int* ei = enc_idx + (size_t)b * SEQ;
  for (int s = 0; s < SEQ; ++s) ob[ei[s]] += pc[s];
}

// ---------------------------------------------------------------------------
// weighted_out = (prob_c * normalized match mask) @ encoded
// ---------------------------------------------------------------------------
__global__ void weighted_kernel(const float* __restrict__ prob_c, const int* __restrict__ enc_idx,
                                const int* __restrict__ input_idx, const float* __restrict__ encoded,
                                float* __restrict__ wout) {
  __shared__ float attn[SEQ];
  __shared__ float red[256];
  const int b = blockIdx.x, tid = threadIdx.x;
  const int ii = input_idx[b];
  const int* ei = enc_idx + (size_t)b * SEQ;
  float part = 0.f;
  for (int s = tid; s < SEQ; s += 256) part += (ei[s] == ii) ? 1.f : 0.f;
  red[tid] = part; __syncthreads();
  for (int off = 128; off > 0; off >>= 1) {
    if (tid < off) red[tid] += red[tid + off];
    __syncthreads();
  }
  const float ssum = red[0];
  const float scale = (ssum > 1.f) ? (1.f / ssum) : 1.f;
  for (int s = tid; s < SEQ; s += 256) {
    const float mv = (ei[s] == ii) ? 1.f : 0.f;
    attn[s] = prob_c[(size_t)b * SEQ + s] * mv * scale;
  }
  __syncthreads();
  const float* eb = encoded + (size_t)b * SEQ * H2;
  for (int h = tid; h < H2; h += 256) {
    float acc = 0.f;
    for (int s = 0; s < SEQ; ++s) acc += attn[s] * eb[(size_t)s * H2 + h];
    wout[(size_t)b * H2 + h] = acc;
  }
}

extern "C" void kernel_launch(void* const* d_in, const int* in_sizes, int n_in,
                              void* d_out, int out_size, void* d_ws, size_t ws_size,
                              hipStream_t stream) {
  const int*   input_idx  = (const int*)  d_in[0];
  const float* encoded    = (const float*)d_in[1];
  const int*   enc_idx    = (const int*)  d_in[2];
  const float* prev_state = (const float*)d_in[3];
  const float* weighted   = (const float*)d_in[4];
  const int*   order      = (const int*)  d_in[5];
  const float* embed_W    = (const float*)d_in[6];
  const float* Wih        = (const float*)d_in[7];
  const float* Whh        = (const float*)d_in[8];
  const float* bih        = (const float*)d_in[9];
  const float* bhh        = (const float*)d_in[10];
  const float* Ws_W       = (const float*)d_in[11];
  const float* Ws_b       = (const float*)d_in[12];
  const float* Wo_W       = (const float*)d_in[13];
  const float* Wo_b       = (const float*)d_in[14];
  const float* Wc_W       = (const float*)d_in[15];
  const float* Wc_b       = (const float*)d_in[16];

  float* out       = (float*)d_out;               // (BS, V)
  float* out_state = out + (size_t)BS * V;        // (BS, H)
  float* out_w     = out_state + (size_t)BS * H;  // (BS, 2H)

  float* ws    = (float*)d_ws;
  float* x     = ws;                   // BS*XW      = 81920
  float* ps    = x + (size_t)BS * XW;  // BS*H       = 32768
  float* gi    = ps + (size_t)BS * H;  // BS*3H      = 98304
  float* gh    = gi + (size_t)BS * H3; // BS*3H      = 98304
  float* state = gh + (size_t)BS * H3; // BS*H       = 32768
  float* part  = state + (size_t)BS*H; // MT*SPLIT*16= 102400
  float* scc   = part + (size_t)MT * SPLIT * 16; // BS*SEQ = 12800

  // 1. prologue: x = [embed, weighted], effective prev_state
  prologue_kernel<<<BS, 256, 0, stream>>>(input_idx, encoded, prev_state, weighted,
                                          order, embed_W, Ws_W, Ws_b, x, ps);
  // 2-3. GRU GEMMs: NTILE=4 -> (64/16)*(96/4) = 96 waves -> 24 blocks of 4 waves
  gemm_bias_wmma<4, XW><<<24, 128, 0, stream>>>(x,  Wih, bih, gi, BS, H3);
  gemm_bias_wmma<4, H ><<<24, 128, 0, stream>>>(ps, Whh, bhh, gh, BS, H3);
  // 4. gates -> state (also written to d_out)
  gates_kernel<<<(BS * H) / 256, 256, 0, stream>>>(gi, gh, ps, state, out_state);
  // 5. score_g: N=50000 -> 3125 col tiles; NTILE=5 -> 4*625 = 2500 waves -> 625 blocks
  gemm_bias_wmma<5, H><<<625, 128, 0, stream>>>(state, Wo_W, Wo_b, out, BS, V);
  // 6-7. fused score_c: 800*8 = 6400 waves -> 1600 blocks
  score_c_gemm<<<(MT * SPLIT) / 4, 128, 0, stream>>>(encoded, Wc_W, Wc_b, state, part);
  score_c_finish<<<(BS * SEQ + 255) / 256, 256, 0, stream>>>(part, enc_idx, scc);
  // 8. softmax over [score_g | score_c] in place
  softmax_kernel<<<BS, 256, 0, stream>>>(out, scc);
  // 9. deterministic scatter-add of copy probs into vocab
  scatter_kernel<<<1, 64, 0, stream>>>(scc, enc_idx, out);
  // 10. attention-weighted encoder sum
  weighted_kernel<<<BS, 256, 0, stream>>>(scc, enc_idx, input_idx, encoded, out_w);
}